// GCNBlock_30202210025887
// MI455X (gfx1250) — compile-verified
//
#include <hip/hip_runtime.h>

#define D 128

typedef float v2f __attribute__((ext_vector_type(2)));
typedef float v8f __attribute__((ext_vector_type(8)));

// ---------------------------------------------------------------------------
// init: deg = 1 (self loop), bn sums = 0
// ---------------------------------------------------------------------------
__global__ void gcn_init_k(float* __restrict__ deg, float* __restrict__ bn, int N) {
    int i = blockIdx.x * blockDim.x + threadIdx.x;
    if (i < N) deg[i] = 1.0f;
    if (i < 2 * D) bn[i] = 0.0f;
}

// ---------------------------------------------------------------------------
// degree: deg[dst] += 1 per edge (hw fp32 atomic, L2-resident)
// ---------------------------------------------------------------------------
__global__ void gcn_degree_k(const int* __restrict__ dstI, float* __restrict__ deg, int E) {
    int i = blockIdx.x * blockDim.x + threadIdx.x;
    if (i < E) unsafeAtomicAdd(&deg[dstI[i]], 1.0f);
}

// ---------------------------------------------------------------------------
// dinv: deg -> rsqrt(deg) in place
// ---------------------------------------------------------------------------
__global__ void gcn_dinv_k(float* __restrict__ deg, int N) {
    int i = blockIdx.x * blockDim.x + threadIdx.x;
    if (i < N) deg[i] = rsqrtf(deg[i]);
}

// ---------------------------------------------------------------------------
// Fused dual GEMM with f32 WMMA (16x16x4):
//   h   = x @ W^T
//   acc = x @ resW^T + bias + dinv^2 * h    (self-loop message folded in)
// One block = 16-row strip of x staged in LDS; 8 waves = 8 column tiles.
// ---------------------------------------------------------------------------
__global__ void __launch_bounds__(256, 2)
gcn_gemm_k(const float* __restrict__ x, const float* __restrict__ W,
           const float* __restrict__ resW, const float* __restrict__ bias,
           const float* __restrict__ dinv,
           float* __restrict__ h, float* __restrict__ acc) {
    __shared__ float xs[16 * 132];  // padded stride 132 floats -> conflict-free

    const int t  = threadIdx.x;
    const int m0 = blockIdx.x * 16;

    // stage 16x128 strip of x into LDS (512 float4s, 256 threads x 2)
    {
        const float4* xg = (const float4*)(x + (size_t)m0 * D);
        float4* xs4 = (float4*)xs;  // 16 rows x 33 float4 (pad = 1)
        int f0 = t, f1 = t + 256;
        xs4[(f0 >> 5) * 33 + (f0 & 31)] = xg[f0];
        xs4[(f1 >> 5) * 33 + (f1 & 31)] = xg[f1];
    }
    __syncthreads();

    const int lane  = t & 31;
    const int n0    = (t >> 5) * 16;        // wave -> column tile
    const int r     = lane & 15;            // A: row, B: column within tile
    const int khalf = (lane >> 4) << 1;     // lanes 0-15: K=0,1; lanes 16-31: K=2,3

    const float* wp = W    + (size_t)(n0 + r) * D + khalf;
    const float* rp = resW + (size_t)(n0 + r) * D + khalf;
    const float* ap = xs + r * 132 + khalf;

    v8f ch = {0.f, 0.f, 0.f, 0.f, 0.f, 0.f, 0.f, 0.f};
    v8f cr = {0.f, 0.f, 0.f, 0.f, 0.f, 0.f, 0.f, 0.f};
#pragma unroll 8
    for (int k = 0; k < D; k += 4) {
        v2f a  = *(const v2f*)(ap + k);
        v2f bw = *(const v2f*)(wp + k);
        v2f br = *(const v2f*)(rp + k);
        ch = __builtin_amdgcn_wmma_f32_16x16x4_f32(false, a, false, bw,
                                                   (short)0, ch, false, false);
        cr = __builtin_amdgcn_wmma_f32_16x16x4_f32(false, a, false, br,
                                                   (short)0, cr, false, false);
    }

    // C/D layout: VGPR j -> row m0 + j (+8 for lanes 16-31), col n0 + (lane&15)
    const int   col  = n0 + r;
    const int   rowb = m0 + ((lane >> 4) << 3);
    const float b    = bias[col];
#pragma unroll
    for (int j = 0; j < 8; ++j) {
        const int   row = rowb + j;
        const float di  = dinv[row];
        const float hv  = ch[j];
        h[(size_t)row * D + col]   = hv;
        acc[(size_t)row * D + col] = cr[j] + b + hv * di * di;
    }
}

// ---------------------------------------------------------------------------
// Edge scatter: one wave per edge; lane owns 4 features.
//   acc[dst] += h[src] * dinv[src]*dinv[dst]
// ---------------------------------------------------------------------------
__global__ void __launch_bounds__(256)
gcn_scatter_k(const int* __restrict__ srcI, const int* __restrict__ dstI,
              const float* __restrict__ dinv, const float* __restrict__ h,
              float* __restrict__ acc, int E) {
    const int lane = threadIdx.x & 31;
    int       wave = (blockIdx.x * blockDim.x + threadIdx.x) >> 5;
    const int nw   = (gridDim.x * blockDim.x) >> 5;
    for (int e = wave; e < E; e += nw) {
        const int    s   = srcI[e];
        const int    d   = dstI[e];
        const float  nrm = dinv[s] * dinv[d];
        const float4 hv  = *(const float4*)(h + (size_t)s * D + lane * 4);
        float*       a   = acc + (size_t)d * D + lane * 4;
        unsafeAtomicAdd(a + 0, hv.x * nrm);
        unsafeAtomicAdd(a + 1, hv.y * nrm);
        unsafeAtomicAdd(a + 2, hv.z * nrm);
        unsafeAtomicAdd(a + 3, hv.w * nrm);
    }
}

// ---------------------------------------------------------------------------
// BN reduce: per-feature sum / sumsq over rows (col = tid & 127, coalesced)
// ---------------------------------------------------------------------------
__global__ void __launch_bounds__(256)
gcn_bnred_k(const float* __restrict__ acc, float* __restrict__ sum,
            float* __restrict__ sumsq, int N) {
    const int col = threadIdx.x & (D - 1);
    int       row = blockIdx.x * (blockDim.x >> 7) + (threadIdx.x >> 7);
    const int rs  = gridDim.x * (blockDim.x >> 7);
    float s = 0.f, q = 0.f;
    for (; row < N; row += rs) {
        float v = acc[(size_t)row * D + col];
        s += v;
        q += v * v;
    }
    unsafeAtomicAdd(&sum[col], s);
    unsafeAtomicAdd(&sumsq[col], q);
}

// ---------------------------------------------------------------------------
// BN finalize: scale = gamma*rsqrt(var+eps), shift = beta - mean*scale
// ---------------------------------------------------------------------------
__global__ void gcn_bnfin_k(const float* __restrict__ sum, const float* __restrict__ sumsq,
                            const float* __restrict__ gamma, const float* __restrict__ beta,
                            float* __restrict__ scale, float* __restrict__ shift, int N) {
    int c = threadIdx.x;
    if (c < D) {
        float invN = 1.0f / (float)N;
        float mean = sum[c] * invN;
        float var  = sumsq[c] * invN - mean * mean;
        float sc   = gamma[c] * rsqrtf(var + 1e-5f);
        scale[c] = sc;
        shift[c] = beta[c] - mean * sc;
    }
}

// ---------------------------------------------------------------------------
// Apply BN + ReLU in place on d_out
// ---------------------------------------------------------------------------
__global__ void __launch_bounds__(256)
gcn_apply_k(float* __restrict__ acc, const float* __restrict__ scale,
            const float* __restrict__ shift, size_t total) {
    size_t i = (size_t)blockIdx.x * blockDim.x + threadIdx.x;
    if (i < total) {
        int   c = (int)(i & (D - 1));
        float v = acc[i] * scale[c] + shift[c];
        acc[i] = v > 0.f ? v : 0.f;
    }
}

// ---------------------------------------------------------------------------
extern "C" void kernel_launch(void* const* d_in, const int* in_sizes, int n_in,
                              void* d_out, int out_size, void* d_ws, size_t ws_size,
                              hipStream_t stream) {
    const float* x     = (const float*)d_in[0];
    const float* W     = (const float*)d_in[1];
    const float* bias  = (const float*)d_in[2];
    const float* resW  = (const float*)d_in[3];
    const float* gamma = (const float*)d_in[4];
    const float* beta  = (const float*)d_in[5];
    const int*   ei    = (const int*)d_in[6];

    const int N = in_sizes[0] / D;   // 50000
    const int E = in_sizes[6] / 2;   // 800000
    const int* srcI = ei;            // edge_index[0]
    const int* dstI = ei + E;        // edge_index[1]

    // workspace layout (floats): deg[N] | h[N*D] | bn(sum,sumsq,scale,shift)[4*D]
    float* ws    = (float*)d_ws;
    float* deg   = ws;
    float* h     = ws + (((size_t)N + 127) & ~(size_t)127);
    float* bn    = h + (size_t)N * D;
    float* bnsum = bn;
    float* bnsq  = bn + D;
    float* scale = bn + 2 * D;
    float* shift = bn + 3 * D;

    float* acc = (float*)d_out;

    const int threads = 256;

    // 1. init deg=1, bn sums=0
    gcn_init_k<<<(N + threads - 1) / threads, threads, 0, stream>>>(deg, bn, N);
    // 2. degree accumulation
    gcn_degree_k<<<(E + threads - 1) / threads, threads, 0, stream>>>(dstI, deg, E);
    // 3. deg -> rsqrt(deg)
    gcn_dinv_k<<<(N + threads - 1) / threads, threads, 0, stream>>>(deg, N);
    // 4. fused WMMA GEMMs + bias + self-loop  (N/16 blocks, 8 col-tile waves each)
    gcn_gemm_k<<<N / 16, threads, 0, stream>>>(x, W, resW, bias, deg, h, acc);
    // 5. edge scatter with hw fp32 atomics
    gcn_scatter_k<<<16384, threads, 0, stream>>>(srcI, dstI, deg, h, acc, E);
    // 6. BN reduction
    gcn_bnred_k<<<512, threads, 0, stream>>>(acc, bnsum, bnsq, N);
    // 7. BN finalize
    gcn_bnfin_k<<<1, D, 0, stream>>>(bnsum, bnsq, gamma, beta, scale, shift, N);
    // 8. normalize + ReLU in place
    size_t total = (size_t)N * D;
    gcn_apply_k<<<(unsigned)((total + threads - 1) / threads), threads, 0, stream>>>(
        acc, scale, shift, total);
}